// TwoOptMCMC_86148454023515
// MI455X (gfx1250) — compile-verified
//
#include <hip/hip_runtime.h>
#include <stdint.h>

// ---------------------------------------------------------------------------
// 2-opt block-move MCMC over permutations (n=4096), T=5722 sequential steps.
//
// Strategy (MI455X):
//  * O(1) delta-weight per step (3 changed edges) => no full 4096-gather,
//    and the running weight w itself is never needed (accept uses only delta).
//  * All proposal RNG is state-independent -> precomputed grid-parallel
//    (device Threefry2x32 faithful to JAX key/split/randint/uniform).
//  * Initial permutation = JAX _shuffle: 36 rounds of stable sort by random
//    u32 keys -> bitonic sort in LDS on (key<<32)|pos composite keys.
//  * Chain runs in ONE workgroup (1024 thr = 32 wave32); permutation in LDS.
//  * Proposal stream prefetched into LDS via CDNA5 async global->LDS loads
//    (global_load_async_to_lds_b128, ASYNCcnt), double-buffered 256-step
//    chunks: per-step param fetch is an LDS broadcast, off the vmem path.
//  * Samples streamed out via CDNA5 async LDS->global stores
//    (global_store_async_from_lds_b128, s_wait_asynccnt), overlapped with
//    the next ~11 chain steps.
// ---------------------------------------------------------------------------

#define NW      4096
#define TSTEPS  (100 + 511 * 11 + 1)   // 5722
#define BLK     1024
#define NSAMP   512
#define CHUNK   256                    // params per prefetch chunk (4 KB)
#define NCHUNK  ((TSTEPS + CHUNK - 1) / CHUNK)   // 23
#define NPAD    (NCHUNK * CHUNK)                 // 5888 padded param slots

// ------------------------------ Threefry-2x32 ------------------------------
__device__ __forceinline__ void tf2x32(uint32_t k0, uint32_t k1,
                                       uint32_t c0, uint32_t c1,
                                       uint32_t& o0, uint32_t& o1) {
  const uint32_t ks2 = 0x1BD11BDAu ^ k0 ^ k1;
  uint32_t x0 = c0 + k0, x1 = c1 + k1;
#define TFR(r) { x0 += x1; x1 = (x1 << (r)) | (x1 >> (32 - (r))); x1 ^= x0; }
  TFR(13) TFR(15) TFR(26) TFR(6)   x0 += k1;  x1 += ks2 + 1u;
  TFR(17) TFR(29) TFR(16) TFR(24)  x0 += ks2; x1 += k0  + 2u;
  TFR(13) TFR(15) TFR(26) TFR(6)   x0 += k0;  x1 += k1  + 3u;
  TFR(17) TFR(29) TFR(16) TFR(24)  x0 += k1;  x1 += ks2 + 4u;
  TFR(13) TFR(15) TFR(26) TFR(6)   x0 += ks2; x1 += k0  + 5u;
#undef TFR
  o0 = x0; o1 = x1;
}

// JAX randint offset in [0, span): double-width remainder trick, u32 wrap.
__device__ __forceinline__ uint32_t randint_off(uint32_t k0, uint32_t k1,
                                                uint32_t span) {
  uint32_t hi, lo;
  tf2x32(k0, k1, 0u, 1u, hi, lo);           // bits shape (2,): [o0, o1]
  uint32_t m = 65536u % span;               // 2^16 mod span
  m = (m * m) % span;                       // 2^32 mod span (u32 wrap as JAX)
  uint32_t off = (hi % span) * m + (lo % span);
  return off % span;
}

// JAX uniform(): bits -> [1,2) -> -1.0
__device__ __forceinline__ float juniform(uint32_t k0, uint32_t k1) {
  uint32_t a, b;
  tf2x32(k0, k1, 0u, 0u, a, b);             // odd-size count padded with 0
  return __uint_as_float((a >> 9) | 0x3F800000u) - 1.0f;
}

// ----------------------- CDNA5 async LDS<->global ops ----------------------
// clang builtin signatures (probed via diagnostics): both take
// (global int4* [AS1], lds int4* [AS3], i32 offset, i32 cpol).
typedef int v4i __attribute__((ext_vector_type(4)));
typedef __attribute__((address_space(1))) v4i gv4i;   // global int4*
typedef __attribute__((address_space(3))) v4i lv4i;   // LDS int4*
typedef __attribute__((address_space(3))) void as3_void;

__device__ __forceinline__ void async_store_b128(void* gptr, void* lptr) {
#if __has_builtin(__builtin_amdgcn_global_store_async_from_lds_b128)
  __builtin_amdgcn_global_store_async_from_lds_b128(
      (gv4i*)gptr, (lv4i*)lptr, 0, 0);
#else
  asm volatile("global_store_async_from_lds_b128 %0, %1, off"
               :: "v"(gptr),
                  "v"((uint32_t)(uintptr_t)(as3_void*)lptr)
               : "memory");
#endif
}

__device__ __forceinline__ void async_load_b128(void* lptr, const void* gptr) {
#if __has_builtin(__builtin_amdgcn_global_load_async_to_lds_b128)
  __builtin_amdgcn_global_load_async_to_lds_b128(
      (gv4i*)gptr, (lv4i*)lptr, 0, 0);
#else
  asm volatile("global_load_async_to_lds_b128 %0, %1, off"
               :: "v"((uint32_t)(uintptr_t)(as3_void*)lptr),
                  "v"(gptr)
               : "memory");
#endif
}

template <int N>
__device__ __forceinline__ void wait_asynccnt() {
#if __has_builtin(__builtin_amdgcn_s_wait_asynccnt)
  __builtin_amdgcn_s_wait_asynccnt(N);
#else
  asm volatile("s_wait_asynccnt %0" :: "i"(N) : "memory");
#endif
}

// ---------------------- kernel 1: proposal precompute ----------------------
// Grid-parallel: every step's (lo,hi,shift,log u) depends only on step keys.
__global__ __launch_bounds__(256) void tom_precompute(int4* __restrict__ params) {
  const int t = blockIdx.x * blockDim.x + threadIdx.x;
  if (t >= NPAD) return;
  if (t >= TSTEPS) { params[t] = make_int4(0, 1, 0, 0); return; }  // inert pad

  // key(42) -> split -> (kinit, kchain); counts split-in-halves: (0,2),(1,3)
  uint32_t a0, b0, a1, b1;
  tf2x32(0u, 42u, 0u, 2u, a0, b0);
  tf2x32(0u, 42u, 1u, 3u, a1, b1);
  const uint32_t kc0 = b0, kc1 = b1;        // kchain = (o1 of both pairs)

  // keys = split(kchain, T): out = [a_0..a_{T-1}, b_0..b_{T-1}];
  // key_t = (out[2t], out[2t+1]); (a_i,b_i) = tf(kchain, (i, T+i)).
  uint32_t e0, e1, x, y;
  if (2 * t + 1 < TSTEPS) {
    tf2x32(kc0, kc1, (uint32_t)(2 * t),     (uint32_t)(TSTEPS + 2 * t),     x, y); e0 = x;
    tf2x32(kc0, kc1, (uint32_t)(2 * t + 1), (uint32_t)(TSTEPS + 2 * t + 1), x, y); e1 = x;
  } else {
    tf2x32(kc0, kc1, (uint32_t)(2 * t     - TSTEPS), (uint32_t)(2 * t),     x, y); e0 = y;
    tf2x32(kc0, kc1, (uint32_t)(2 * t + 1 - TSTEPS), (uint32_t)(2 * t + 1), x, y); e1 = y;
  }

  // split(k_t, 4): counts (0,4),(1,5),(2,6),(3,7);
  // kl=(a0,a1) ka=(a2,a3) kr=(b0,b1) ku=(b2,b3)
  uint32_t aA, bA, aB, bB, aC, bC, aD, bD;
  tf2x32(e0, e1, 0u, 4u, aA, bA);
  tf2x32(e0, e1, 1u, 5u, aB, bB);
  tf2x32(e0, e1, 2u, 6u, aC, bC);
  tf2x32(e0, e1, 3u, 7u, aD, bD);

  const uint32_t l = 1u + randint_off(aA, aB, (uint32_t)(NW - 1));   // [1, n-1]
  const uint32_t a = randint_off(aC, aD, (uint32_t)(NW - l + 1));    // [0, n-l]
  const uint32_t b = a + l;
  const uint32_t r = randint_off(bA, bB, (uint32_t)(NW - l));        // [0, n-l-1]
  uint32_t c = b + r + 1u;
  if (c >= (uint32_t)(NW + 1)) c -= (uint32_t)(NW + 1);              // mod n+1

  int lo, hi, shift;
  if (a < c) { lo = (int)a; hi = (int)c; shift = (int)(b - a); }
  else       { lo = (int)c; hi = (int)b; shift = (int)(a - c); }

  const float logu = logf(juniform(bC, bD));
  params[t] = make_int4(lo, hi, shift, __float_as_int(logu));
}

// ----------------------- kernel 2: sequential chain ------------------------
__global__ __launch_bounds__(BLK) void tom_chain(const float* __restrict__ bigram,
                                                 const float* __restrict__ startw,
                                                 const float* __restrict__ endw,
                                                 const int4*  __restrict__ params,
                                                 int* __restrict__ out) {
  __shared__ __align__(16) int sh_p[NW];                 // 16 KB permutation
  // 32 KB multi-use: sort keys during init; then [0,16KB) sample staging,
  // [16KB,24KB) two 4KB async param chunks.
  __shared__ __align__(16) unsigned long long sh_k[NW];
  __shared__ float sh_red[8];
  __shared__ int   sh_flag;
  const int tid = threadIdx.x;

  // ---- initial permutation: JAX _shuffle(kinit, arange(4096)), 36 rounds ----
  uint32_t a0, b0, a1, b1;
  tf2x32(0u, 42u, 0u, 2u, a0, b0);
  tf2x32(0u, 42u, 1u, 3u, a1, b1);
  uint32_t ik0 = a0, ik1 = a1;              // kinit

  for (int i = tid; i < NW; i += BLK) sh_p[i] = i;
  __syncthreads();

  for (int rnd = 0; rnd < 36; ++rnd) {
    // kinit, sub = split(kinit)
    uint32_t s0a, s0b, s1a, s1b;
    tf2x32(ik0, ik1, 0u, 2u, s0a, s0b);
    tf2x32(ik0, ik1, 1u, 3u, s1a, s1b);
    const uint32_t sub0 = s0b, sub1 = s1b;
    ik0 = s0a; ik1 = s1a;

    // sort_keys = random_bits(sub, 32, (4096,)): halves counter layout
    for (int i = tid; i < NW; i += BLK) {
      uint32_t u, v, bits;
      if (i < NW / 2) { tf2x32(sub0, sub1, (uint32_t)i, (uint32_t)(NW / 2 + i), u, v); bits = u; }
      else            { tf2x32(sub0, sub1, (uint32_t)(i - NW / 2), (uint32_t)i, u, v); bits = v; }
      sh_k[i] = ((unsigned long long)bits << 32) | (unsigned)i;   // stable composite
    }
    __syncthreads();

    // bitonic ascending sort of composite keys (== stable sort by bits)
    for (int k = 2; k <= NW; k <<= 1) {
      for (int j = k >> 1; j > 0; j >>= 1) {
        for (int i = tid; i < NW; i += BLK) {
          const int ixj = i ^ j;
          if (ixj > i) {
            const unsigned long long A = sh_k[i], B = sh_k[ixj];
            const bool up = ((i & k) == 0);
            if ((A > B) == up) { sh_k[i] = B; sh_k[ixj] = A; }
          }
        }
        __syncthreads();
      }
    }

    // apply: p_new[i] = p_old[orig_pos(sorted key i)]
    int tmp[NW / BLK];
    { int m = 0; for (int i = tid; i < NW; i += BLK, ++m) tmp[m] = sh_p[(int)(sh_k[i] & 0xFFFFu)]; }
    __syncthreads();
    { int m = 0; for (int i = tid; i < NW; i += BLK, ++m) sh_p[i] = tmp[m]; }
    __syncthreads();
  }

  // ------------------------------- chain -----------------------------------
  int*  stage     = (int*)sh_k;                       // 4096 ints (16 KB)
  int4* parms_lds = (int4*)(stage + NW);              // 2 x CHUNK int4 (8 KB)

  // pre-issue param chunks 0 and 1 (ASYNCcnt-tracked global->LDS DMA)
  if (tid < CHUNK) {
    async_load_b128(parms_lds + tid,         params + tid);
    async_load_b128(parms_lds + CHUNK + tid, params + CHUNK + tid);
  }

  for (int t = 0; t < TSTEPS; ++t) {
    const int c = t >> 8;                   // current chunk
    if ((t & (CHUNK - 1)) == 0) {
      wait_asynccnt<0>();                   // chunk c resident in LDS
      __syncthreads();
      const int nc = c + 1;
      if (c >= 1 && nc < NCHUNK && tid < CHUNK)   // refill buffer just freed
        async_load_b128(parms_lds + (nc & 1) * CHUNK + tid,
                        params + nc * CHUNK + tid);
    }

    const int4 pr = parms_lds[(c & 1) * CHUNK + (t & (CHUNK - 1))];
    const int lo = pr.x, hi = pr.y, shift = pr.z;
    const float logu = __int_as_float(pr.w);
    const int span = hi - lo;

    if (shift != 0) {                        // shift==0 => identity rotation, no-op
      // 3 changed edges: 6 parallel gathers (one per lane), fixed-order sum.
      if (tid < 6) {
        float v = 0.0f;
        if (tid == 0) v =  (lo > 0)  ? bigram[(size_t)sh_p[lo - 1] * NW + sh_p[lo + shift]]
                                     : startw[sh_p[lo + shift]];
        if (tid == 1) v = -((lo > 0) ? bigram[(size_t)sh_p[lo - 1] * NW + sh_p[lo]]
                                     : startw[sh_p[lo]]);
        if (tid == 2) v =  bigram[(size_t)sh_p[hi - 1] * NW + sh_p[lo]];
        if (tid == 3) v = -bigram[(size_t)sh_p[lo + shift - 1] * NW + sh_p[lo + shift]];
        if (tid == 4) v =  (hi < NW)  ? bigram[(size_t)sh_p[lo + shift - 1] * NW + sh_p[hi]]
                                      : endw[sh_p[lo + shift - 1]];
        if (tid == 5) v = -((hi < NW) ? bigram[(size_t)sh_p[hi - 1] * NW + sh_p[hi]]
                                      : endw[sh_p[hi - 1]]);
        sh_red[tid] = v;
      }
      __syncthreads();
      if (tid == 0) {
        const float d = ((sh_red[0] + sh_red[1]) + (sh_red[2] + sh_red[3])) + (sh_red[4] + sh_red[5]);
        sh_flag = (fminf(d, 0.0f) > logu) ? 1 : 0;   // accept test (w2-w == delta)
      }
      __syncthreads();

      if (sh_flag) {                         // commit rotation in LDS
        int nv[NW / BLK]; int m = 0;
        for (int i = lo + tid; i < hi; i += BLK, ++m) {
          int jj = i - lo + shift; if (jj >= span) jj -= span;
          nv[m] = sh_p[lo + jj];
        }
        __syncthreads();
        m = 0;
        for (int i = lo + tid; i < hi; i += BLK, ++m) sh_p[i] = nv[m];
        __syncthreads();
      }
    }

    // ------- sampling: perms[100::11][:512], async-streamed to global -------
    if (t >= 100 && (t - 100) % 11 == 0) {
      const int s = (t - 100) / 11;
      if (s < NSAMP) {
        wait_asynccnt<0>();                  // prior store from staging done
        __syncthreads();                     // ... across all waves
        ((v4i*)stage)[tid] = ((const v4i*)sh_p)[tid];   // lane-contiguous copy
        __syncthreads();                     // LDS writes visible/complete
        // one b128 per lane: 1024 lanes x 16B == whole 16KB sample; store
        // overlaps the next ~11 chain steps.
        async_store_b128(out + (size_t)s * NW + tid * 4, stage + tid * 4);
      }
    }
  }

  wait_asynccnt<0>();                        // drain async stores before exit
}

// ------------------------------ launch glue --------------------------------
extern "C" void kernel_launch(void* const* d_in, const int* in_sizes, int n_in,
                              void* d_out, int out_size, void* d_ws, size_t ws_size,
                              hipStream_t stream) {
  (void)in_sizes; (void)n_in; (void)out_size; (void)ws_size;
  // d_in[0] = n_words (scalar, ==4096), d_in[1]=bigram, d_in[2]=start, d_in[3]=end
  const float* bigram = (const float*)d_in[1];
  const float* startw = (const float*)d_in[2];
  const float* endw   = (const float*)d_in[3];
  int4* params = (int4*)d_ws;                // NPAD * 16B = ~94 KB scratch

  tom_precompute<<<(NPAD + 255) / 256, 256, 0, stream>>>(params);
  tom_chain<<<1, BLK, 0, stream>>>(bigram, startw, endw, params, (int*)d_out);
}